// WRGAT_62689342652855
// MI455X (gfx1250) — compile-verified
//
#include <hip/hip_runtime.h>
#include <math.h>

// Problem constants (match reference setup_inputs()).
#define RREL 8
#define NNODE 50000
#define EEDGE 1600000
#define FIN 256
#define DIMS1 64
#define NCLS 16

typedef __attribute__((ext_vector_type(2))) float v2f;
typedef __attribute__((ext_vector_type(8))) float v8f;

// ---------------------------------------------------------------------------
// Order-preserving float <-> uint mapping so we can use integer atomicMax for
// the per-(relation,dst) segment max of attention scores.
// ---------------------------------------------------------------------------
__device__ __forceinline__ unsigned f2o(float f) {
    unsigned u = __float_as_uint(f);
    return (u & 0x80000000u) ? ~u : (u | 0x80000000u);
}
__device__ __forceinline__ float o2f(unsigned o) {
    unsigned u = (o & 0x80000000u) ? (o ^ 0x80000000u) : ~o;
    return __uint_as_float(u);
}

// ---------------------------------------------------------------------------
// Dense projection via fp32 WMMA (V_WMMA_F32_16X16X4_F32).
// One block = 9 waves (wave32). The 16xK row-tile of x is staged in LDS once;
// wave w then computes the full 16xD tile of  out[w] = x_tile @ B_w  where
// B_w = W[w] for w<RREL, or root (+bias) for w==RREL.
//   A (16x4 fp32): lane l -> row M=l%16, ks=(l/16)*2; a={A[M][k+ks],A[M][k+ks+1]}
//   B (4x16 fp32): lane l -> col N=l%16, ks=(l/16)*2; b={B[k+ks][N],B[k+ks+1][N]}
//   D (16x16):     lane l -> col N=l%16, row (l/16)*8 + v for acc element v.
// LDS rows padded by 2 dwords -> conflict-free ds_load_b64 A-fragment reads.
// N = 50000 is a multiple of 16: no guards, EXEC all-ones at every WMMA.
// ---------------------------------------------------------------------------
template<int K, int NT>          // D = NT*16 output columns
__global__ __launch_bounds__(288)
void proj_wmma(const float* __restrict__ x,      // [N,K]
               const float* __restrict__ W,      // [RREL,K,D]
               const float* __restrict__ root,   // [K,D]
               const float* __restrict__ bias,   // [D]
               float* __restrict__ xr,           // [RREL,N,D]
               float* __restrict__ hinit,        // [N,D]
               int N)
{
    constexpr int D  = NT * 16;
    constexpr int KP = K + 2;                    // padded LDS row stride
    __shared__ float tile[16 * KP];

    const int tid  = threadIdx.x;
    const int lane = tid & 31;
    const int wv   = tid >> 5;                   // 0..8 : matrix index
    const int row0 = blockIdx.x * 16;

    // Stage the 16xK A-tile (contiguous region of x) into LDS.
    const float* __restrict__ xsrc = x + (size_t)row0 * K;
    for (int i = tid; i < 16 * K; i += 288) {
        const int r = i / K, c = i % K;          // K is a power of two
        tile[r * KP + c] = xsrc[i];
    }
    __syncthreads();

    const float* __restrict__ B =
        (wv < RREL) ? (W + (size_t)wv * K * D) : root;
    const int nlo   = lane & 15;
    const int khalf = (lane >> 4) << 1;          // 0 or 2
    const float* __restrict__ arow = tile + nlo * KP + khalf;

    v8f acc[NT];
    #pragma unroll
    for (int j = 0; j < NT; ++j) acc[j] = (v8f){};

    for (int k = 0; k < K; k += 4) {
        v2f a;
        const float2 av = *(const float2*)(arow + k);      // LDS b64 load
        a.x = av.x; a.y = av.y;
        const float* __restrict__ b0 = B + (size_t)(k + khalf) * D + nlo;
        #pragma unroll
        for (int j = 0; j < NT; ++j) {
            v2f b;
            b.x = b0[j * 16];
            b.y = b0[j * 16 + D];
            acc[j] = __builtin_amdgcn_wmma_f32_16x16x4_f32(
                /*neg_a=*/false, a, /*neg_b=*/false, b,
                /*c_mod=*/(short)0, acc[j], /*reuse_a=*/false, /*reuse_b=*/false);
        }
    }

    const int rbase = row0 + ((lane >> 4) << 3);
    if (wv < RREL) {
        float* __restrict__ o = xr + (size_t)wv * N * D;
        #pragma unroll
        for (int j = 0; j < NT; ++j)
            #pragma unroll
            for (int v = 0; v < 8; ++v)
                o[(size_t)(rbase + v) * D + j * 16 + nlo] = acc[j][v];
    } else {
        #pragma unroll
        for (int j = 0; j < NT; ++j) {
            const float bv = bias[j * 16 + nlo];
            #pragma unroll
            for (int v = 0; v < 8; ++v)
                hinit[(size_t)(rbase + v) * D + j * 16 + nlo] = acc[j][v] + bv;
        }
    }
}

// ---------------------------------------------------------------------------
// Edge-count per (relation,dst) segment (shared by both layers).
// ---------------------------------------------------------------------------
__global__ void cnt_kernel(const int* __restrict__ dst,
                           const int* __restrict__ ec,
                           float* __restrict__ cnt, int E, int N)
{
    int e = blockIdx.x * blockDim.x + threadIdx.x;
    if (e >= E) return;
    atomicAdd(&cnt[(size_t)ec[e] * N + dst[e]], 1.0f);
}

// s[e] = leaky_relu( a_i.x_i + a_j.x_j , 0.2 );  atomicMax of segment max.
// D is a multiple of 4 and every row is >=64B aligned -> float4 gathers.
template<int D>
__global__ void edge_score(const float* __restrict__ xr,    // [R,N,D]
                           const int* __restrict__ src,
                           const int* __restrict__ dst,
                           const int* __restrict__ ec,
                           const float* __restrict__ atten, // [R,2D]
                           float* __restrict__ s,
                           unsigned* __restrict__ m,        // [R*N], memset 0
                           int E, int N)
{
    int e = blockIdx.x * blockDim.x + threadIdx.x;
    if (e >= E) return;
    const int r = ec[e], si = src[e], di = dst[e];
    const float4* __restrict__ xi = (const float4*)(xr + ((size_t)r * N + di) * D);
    const float4* __restrict__ xj = (const float4*)(xr + ((size_t)r * N + si) * D);
    const float4* __restrict__ ai = (const float4*)(atten + (size_t)r * 2 * D);
    const float4* __restrict__ aj = (const float4*)(atten + (size_t)r * 2 * D + D);
    float acc = 0.0f;
    #pragma unroll
    for (int q = 0; q < D / 4; ++q) {
        const float4 a = ai[q], u = xi[q], b = aj[q], w = xj[q];
        acc += a.x * u.x + a.y * u.y + a.z * u.z + a.w * u.w
             + b.x * w.x + b.y * w.y + b.z * w.z + b.w * w.w;
    }
    const float sv = (acc > 0.0f) ? acc : 0.2f * acc;
    s[e] = sv;
    atomicMax(&m[(size_t)r * N + di], f2o(sv));
}

// s[e] <- exp(s[e] - m[seg]);  denom[seg] += s[e]
__global__ void edge_exp(float* __restrict__ s,
                         const unsigned* __restrict__ m,
                         float* __restrict__ denom,          // memset 0
                         const int* __restrict__ dst,
                         const int* __restrict__ ec,
                         int E, int N)
{
    int e = blockIdx.x * blockDim.x + threadIdx.x;
    if (e >= E) return;
    const size_t seg = (size_t)ec[e] * N + dst[e];
    const float ev = expf(s[e] - o2f(m[seg]));
    s[e] = ev;
    atomicAdd(&denom[seg], ev);
}

// s[e] <- softmax_alpha * edge_weight / max(cnt,1)   (final scatter coeff)
__global__ void edge_coeff(float* __restrict__ s,
                           const float* __restrict__ denom,
                           const float* __restrict__ cnt,
                           const float* __restrict__ ew,
                           const int* __restrict__ dst,
                           const int* __restrict__ ec,
                           int E, int N)
{
    int e = blockIdx.x * blockDim.x + threadIdx.x;
    if (e >= E) return;
    const size_t seg = (size_t)ec[e] * N + dst[e];
    const float alpha = s[e] / fmaxf(denom[seg], 1e-16f);
    s[e] = alpha * ew[e] / fmaxf(cnt[seg], 1.0f);
}

// out[dst, 4q..4q+3] += coeff[e] * xr[ec, src, 4q..4q+3]
// One thread per (edge, float4 chunk): b128 gather + 4 global_atomic_add_f32.
template<int D>
__global__ void edge_scatter(const float* __restrict__ xr,   // [R,N,D]
                             const float* __restrict__ coeff,
                             const int* __restrict__ src,
                             const int* __restrict__ dst,
                             const int* __restrict__ ec,
                             float* __restrict__ out,        // [N,D]
                             long long total, int N)         // total = E*D/4
{
    constexpr int Q = D / 4;
    long long t = (long long)blockIdx.x * blockDim.x + threadIdx.x;
    if (t >= total) return;
    const int e = (int)(t / Q);
    const int q = (int)(t % Q);
    const float c = coeff[e];
    const float4 v =
        ((const float4*)(xr + ((size_t)ec[e] * N + src[e]) * D))[q];
    float* o = out + (size_t)dst[e] * D + q * 4;
    atomicAdd(o + 0, c * v.x);
    atomicAdd(o + 1, c * v.y);
    atomicAdd(o + 2, c * v.z);
    atomicAdd(o + 3, c * v.w);
}

__global__ void relu_kernel(float* __restrict__ h, long long n)
{
    long long i = (long long)blockIdx.x * blockDim.x + threadIdx.x;
    if (i < n) h[i] = fmaxf(h[i], 0.0f);
}

// In-place log_softmax over NCLS classes, one thread per node.
__global__ void logsoftmax_kernel(float* __restrict__ out, int N)
{
    int n = blockIdx.x * blockDim.x + threadIdx.x;
    if (n >= N) return;
    float* row = out + (size_t)n * NCLS;
    float4 r4[NCLS / 4];
    #pragma unroll
    for (int q = 0; q < NCLS / 4; ++q) r4[q] = ((const float4*)row)[q];
    const float* rf = (const float*)r4;
    float mx = rf[0];
    #pragma unroll
    for (int k = 1; k < NCLS; ++k) mx = fmaxf(mx, rf[k]);
    float sum = 0.0f;
    #pragma unroll
    for (int k = 0; k < NCLS; ++k) sum += expf(rf[k] - mx);
    const float ls = mx + logf(sum);
    #pragma unroll
    for (int q = 0; q < NCLS / 4; ++q) {
        float4 w = r4[q];
        w.x -= ls; w.y -= ls; w.z -= ls; w.w -= ls;
        ((float4*)row)[q] = w;
    }
}

// ---------------------------------------------------------------------------
extern "C" void kernel_launch(void* const* d_in, const int* in_sizes, int n_in,
                              void* d_out, int out_size, void* d_ws, size_t ws_size,
                              hipStream_t stream)
{
    const float* x      = (const float*)d_in[0];
    const int*   eidx   = (const int*)  d_in[1];
    const float* ew     = (const float*)d_in[2];
    const int*   ec     = (const int*)  d_in[3];
    const float* W1     = (const float*)d_in[4];
    const float* atten1 = (const float*)d_in[5];
    const float* root1  = (const float*)d_in[6];
    const float* bias1  = (const float*)d_in[7];
    const float* W2     = (const float*)d_in[8];
    const float* atten2 = (const float*)d_in[9];
    const float* root2  = (const float*)d_in[10];
    const float* bias2  = (const float*)d_in[11];
    float* out = (float*)d_out;

    const int N = NNODE, E = EEDGE;
    const int* src = eidx;         // edge_index[0]
    const int* dst = eidx + E;     // edge_index[1]

    // Workspace carve-up (floats): xr1 | h | xr2 | s | m | denom | cnt
    float*    xr1   = (float*)d_ws;                           // R*N*64
    float*    h     = xr1 + (size_t)RREL * N * DIMS1;         // N*64
    float*    xr2   = h   + (size_t)N * DIMS1;                // R*N*16
    float*    sbuf  = xr2 + (size_t)RREL * N * NCLS;          // E
    unsigned* mbuf  = (unsigned*)(sbuf + E);                  // R*N
    float*    denom = (float*)(mbuf + (size_t)RREL * N);      // R*N
    float*    cnt   = denom + (size_t)RREL * N;               // R*N

    const int TB = 256;
    const int egrid = (E + TB - 1) / TB;

    // Per-(relation,dst) counts: shared by both layers.
    hipMemsetAsync(cnt, 0, (size_t)RREL * N * sizeof(float), stream);
    cnt_kernel<<<egrid, TB, 0, stream>>>(dst, ec, cnt, E, N);

    // ---------------- Layer 1 (K=256, D=64) ----------------
    {
        proj_wmma<FIN, DIMS1 / 16><<<N / 16, 288, 0, stream>>>(
            x, W1, root1, bias1, xr1, h, N);

        hipMemsetAsync(mbuf, 0, (size_t)RREL * N * sizeof(unsigned), stream);
        hipMemsetAsync(denom, 0, (size_t)RREL * N * sizeof(float), stream);

        edge_score<DIMS1><<<egrid, TB, 0, stream>>>(xr1, src, dst, ec, atten1,
                                                    sbuf, mbuf, E, N);
        edge_exp<<<egrid, TB, 0, stream>>>(sbuf, mbuf, denom, dst, ec, E, N);
        edge_coeff<<<egrid, TB, 0, stream>>>(sbuf, denom, cnt, ew, dst, ec, E, N);

        const long long tot = (long long)E * (DIMS1 / 4);
        edge_scatter<DIMS1><<<(unsigned)((tot + TB - 1) / TB), TB, 0, stream>>>(
            xr1, sbuf, src, dst, ec, h, tot, N);

        const long long hn = (long long)N * DIMS1;
        relu_kernel<<<(unsigned)((hn + TB - 1) / TB), TB, 0, stream>>>(h, hn);
    }

    // ---------------- Layer 2 (K=64, D=16) ----------------
    {
        proj_wmma<DIMS1, NCLS / 16><<<N / 16, 288, 0, stream>>>(
            h, W2, root2, bias2, xr2, out, N);

        hipMemsetAsync(mbuf, 0, (size_t)RREL * N * sizeof(unsigned), stream);
        hipMemsetAsync(denom, 0, (size_t)RREL * N * sizeof(float), stream);

        edge_score<NCLS><<<egrid, TB, 0, stream>>>(xr2, src, dst, ec, atten2,
                                                   sbuf, mbuf, E, N);
        edge_exp<<<egrid, TB, 0, stream>>>(sbuf, mbuf, denom, dst, ec, E, N);
        edge_coeff<<<egrid, TB, 0, stream>>>(sbuf, denom, cnt, ew, dst, ec, E, N);

        const long long tot = (long long)E * (NCLS / 4);
        edge_scatter<NCLS><<<(unsigned)((tot + TB - 1) / TB), TB, 0, stream>>>(
            xr2, sbuf, src, dst, ec, out, tot, N);

        logsoftmax_kernel<<<(N + TB - 1) / TB, TB, 0, stream>>>(out, N);
    }
}